// SymmetricContraction_18339510354334
// MI455X (gfx1250) — compile-verified
//
#include <hip/hip_runtime.h>

typedef __attribute__((ext_vector_type(2))) float v2f;
typedef __attribute__((ext_vector_type(8))) float v8f;
typedef __attribute__((ext_vector_type(4))) unsigned int u32x4;
typedef __attribute__((ext_vector_type(4))) int i32x4;
typedef __attribute__((ext_vector_type(8))) int i32x8;

#define D3K 4096   // K for order-3 GEMM = 16^3 (a,b,j)
#define R3  128    // padded rows (119 valid: 8*1 + 12*3 + 15*5)
#define D2K 256    // K for order-2 GEMM = 16^2 (a,b)
#define R2  32     // rows (3*1 + 3*3 + 4*5)
#define D1K 16     // K for order-1 GEMM
#define R1  16     // padded rows (9 valid)
#define KS  32     // K-slab staged into LDS for order-3 A operand (16 KB)
#define NS3 (D3K / KS)

#if defined(__has_builtin)
#if __has_builtin(__builtin_amdgcn_tensor_load_to_lds) && __has_builtin(__builtin_amdgcn_s_wait_tensorcnt)
#define HAVE_TDM 1
#endif
#endif
#ifndef HAVE_TDM
#define HAVE_TDM 0
#endif

// -------------------------------------------------------------------------
// Pack the reshaped basis tensors into K-pair-interleaved layout so an A
// fragment (16x4 f32 WMMA A operand: lane L holds M=L%16, K = v + 2*(L/16))
// is a single b64 load:  P[(k>>1)*2R + 2r + (k&1)]
// -------------------------------------------------------------------------
__global__ __launch_bounds__(256)
void pack_bases_kernel(const float* __restrict__ b30, const float* __restrict__ b31,
                       const float* __restrict__ b32, const float* __restrict__ b20,
                       const float* __restrict__ b21, const float* __restrict__ b22,
                       const float* __restrict__ b10, const float* __restrict__ b11,
                       const float* __restrict__ b12, float* __restrict__ ws) {
  int idx = blockIdx.x * 256 + threadIdx.x;
  float* B3p = ws;
  float* B2p = ws + D3K * R3;
  float* B1p = B2p + D2K * R2;
  if (idx < D3K * R3) {
    int k = idx >> 7, r = idx & 127;            // k = a*256+b*16+j
    float v = 0.0f;
    if (r < 8)        v = b30[k * 8  + r];          // (D,D,D, 8,1)
    else if (r < 44)  v = b31[k * 36 + (r - 8)];    // (D,D,D,12,3)
    else if (r < 119) v = b32[k * 75 + (r - 44)];   // (D,D,D,15,5)
    B3p[(k >> 1) * (2 * R3) + 2 * r + (k & 1)] = v;
  } else if (idx < D3K * R3 + D2K * R2) {
    int j = idx - D3K * R3;
    int k = j >> 5, r = j & 31;                 // k = a*16+b
    float v;
    if (r < 3)       v = b20[k * 3  + r];           // (D,D,3,1)
    else if (r < 12) v = b21[k * 9  + (r - 3)];     // (D,D,3,3)
    else             v = b22[k * 20 + (r - 12)];    // (D,D,4,5)
    B2p[(k >> 1) * (2 * R2) + 2 * r + (k & 1)] = v;
  } else if (idx < D3K * R3 + D2K * R2 + D1K * R1) {
    int j = idx - (D3K * R3 + D2K * R2);
    int k = j >> 4, r = j & 15;
    float v = 0.0f;
    if (r < 1)      v = b10[k];                     // (D,1,1)
    else if (r < 4) v = b11[k * 3 + (r - 1)];       // (D,1,3)
    else if (r < 9) v = b12[k * 5 + (r - 4)];       // (D,1,5)
    B1p[(k >> 1) * (2 * R1) + 2 * r + (k & 1)] = v;
  }
}

#if HAVE_TDM
// LDS byte offset of a __shared__ object (addrspace(3) offset)
__device__ static inline unsigned lds_addr_of(const void* p) {
  return (unsigned)(unsigned long long)(__attribute__((address_space(3))) const char*)p;
}

// Issue a TDM DMA of `nelem` f32 (contiguous) from global -> LDS.
// D# per CDNA5 ISA ch.8: 1D tile (tensor_dim0 = tile_dim0 = nelem, dim1 = 1).
__device__ static inline void tdm_load_slab(const float* __restrict__ gsrc,
                                            unsigned lds_byte, unsigned nelem) {
  unsigned long long ga = (unsigned long long)gsrc;
  u32x4 g0;
  g0[0] = 1u;                                                  // count=1 (valid, user)
  g0[1] = lds_byte;                                            // lds_addr
  g0[2] = (unsigned)ga;                                        // global_addr[31:0]
  g0[3] = (unsigned)((ga >> 32) & 0x1FFFFFFull) | (2u << 30);  // addr[56:32] | type=2
  i32x8 g1;
  g1[0] = (int)(2u << 16);                 // data_size = 4B; wg_mask=0; no pad/iter
  g1[1] = (int)((nelem & 0xFFFFu) << 16);  // tensor_dim0[15:0]; atomic_bar_addr=0
  g1[2] = (int)((nelem >> 16) | (1u << 16));   // tensor_dim0[31:16]; tensor_dim1=1
  g1[3] = (int)((nelem & 0xFFFFu) << 16);  // tensor_dim1[31:16]=0; tile_dim0=nelem
  g1[4] = 1;                               // tile_dim1=1; tile_dim2=0
  g1[5] = (int)nelem;                      // tensor_dim0_stride[31:0]
  g1[6] = 0;                               // stride hi / tensor_dim1_stride lo
  g1[7] = 0;                               // tensor_dim1_stride hi
  i32x4 gz4 = {0, 0, 0, 0};                // groups 2/3 unused (dims 2+ zero)
  i32x8 gz8 = {0, 0, 0, 0, 0, 0, 0, 0};    // extra group (clang-23 6-arg form)
  __builtin_amdgcn_tensor_load_to_lds(g0, g1, gz4, gz4, gz8, 0);
}
#endif

// -------------------------------------------------------------------------
// Main fused kernel: one block per node n (t = type[n] is block-uniform).
// 8 waves x 16 channels; per wave: S3 [128x16] = B3 @ (x (x) x (x) x),
// S2 [32x16], S1 [16x16] via V_WMMA_F32_16X16X4_F32; order-3 A operand is
// double-buffered through LDS via the Tensor Data Mover.
// -------------------------------------------------------------------------
__global__ __launch_bounds__(256)
void mace_symcon_kernel(const float* __restrict__ x, const int* __restrict__ input_type,
                        const float* __restrict__ w30, const float* __restrict__ w31,
                        const float* __restrict__ w32, const float* __restrict__ w20,
                        const float* __restrict__ w21, const float* __restrict__ w22,
                        const float* __restrict__ w10, const float* __restrict__ w11,
                        const float* __restrict__ w12,
                        const float* __restrict__ ws, float* __restrict__ out) {
  __shared__ __align__(16) float lds_x[2048];        // 128 channels * 16 (8 KB)
  __shared__ __align__(16) float lds_A[2][KS * R3];  // double-buffered A slab (2x16 KB)

  const int n    = blockIdx.x;
  const int tid  = threadIdx.x;
  const int wv   = tid >> 5;     // wave id 0..7  -> channel tile
  const int lane = tid & 31;
  const int col  = lane & 15;    // column within 16-channel tile
  const int h    = lane >> 4;    // half-wave

  const float* B3p = ws;
  const float* B2p = ws + D3K * R3;
  const float* B1p = B2p + D2K * R2;

#if HAVE_TDM
  // kick off DMA of slab 0 while we stage x
  if (wv == 0) tdm_load_slab(B3p, lds_addr_of(&lds_A[0][0]), KS * R3);
#endif

  // stage x[n,:,:] (2048 floats)
  {
    const float4* src = (const float4*)(x + (size_t)n * 2048);
    float4* dst = (float4*)lds_x;
    #pragma unroll
    for (int i = 0; i < 2; ++i) dst[tid + i * 256] = src[tid + i * 256];
  }
  __syncthreads();

  const int xbase = wv * 256 + col * 16;   // this lane's channel row of x

  // ---------------- order-3 GEMM: acc3[r,col] over K = 4096 ----------------
  v8f acc3[8] = {};
  for (int s = 0; s < NS3; ++s) {
#if HAVE_TDM
    if (wv == 0) {
      if (s + 1 < NS3) {   // prefetch next slab into the other buffer
        tdm_load_slab(B3p + (size_t)(s + 1) * (KS * R3),
                      lds_addr_of(&lds_A[(s + 1) & 1][0]), KS * R3);
        __builtin_amdgcn_s_wait_tensorcnt(1);   // in-order: slab s has landed
      } else {
        __builtin_amdgcn_s_wait_tensorcnt(0);
      }
    }
    __syncthreads();
#else
    __syncthreads();
    {
      const float4* src = (const float4*)(B3p + (size_t)s * (KS * R3));
      float4* dst = (float4*)&lds_A[s & 1][0];
      #pragma unroll
      for (int i = 0; i < 4; ++i) dst[tid + i * 256] = src[tid + i * 256];
    }
    __syncthreads();
#endif
    const float* Abuf = &lds_A[s & 1][0];
    const int k0 = s * KS;
    #pragma unroll
    for (int kc = 0; kc < KS / 4; ++kc) {
      const int k  = k0 + kc * 4;         // wave-uniform chunk base
      const int ai = k >> 8;
      const int bi = (k >> 4) & 15;
      const int j0 = k & 15;              // in {0,4,8,12}
      const float xab = lds_x[xbase + ai] * lds_x[xbase + bi];
      v2f bf;                              // B operand: K row = 2*v + h
      bf.x = xab * lds_x[xbase + j0 + h];
      bf.y = xab * lds_x[xbase + j0 + 2 + h];
      const int abase = (kc * 2 + h) * (2 * R3);
      #pragma unroll
      for (int t3 = 0; t3 < 8; ++t3) {
        v2f af = *(const v2f*)&Abuf[abase + (t3 * 16 + col) * 2];
        acc3[t3] = __builtin_amdgcn_wmma_f32_16x16x4_f32(
            false, af, false, bf, (short)0, acc3[t3], false, false);
      }
    }
    __syncthreads();   // all waves done with this buffer before it is refilled
  }

  // ---------------- order-2 GEMM: K = 256 (A direct from L2) ---------------
  v8f acc2[2] = {};
  #pragma unroll 4
  for (int kc = 0; kc < 64; ++kc) {
    const int k  = kc * 4;
    const int ai = k >> 4;
    const int b0 = k & 15;
    const float xa = lds_x[xbase + ai];
    v2f bf;
    bf.x = xa * lds_x[xbase + b0 + h];
    bf.y = xa * lds_x[xbase + b0 + 2 + h];
    const int abase = (kc * 2 + h) * (2 * R2);
    #pragma unroll
    for (int t2 = 0; t2 < 2; ++t2) {
      v2f af = *(const v2f*)&B2p[abase + (t2 * 16 + col) * 2];
      acc2[t2] = __builtin_amdgcn_wmma_f32_16x16x4_f32(
          false, af, false, bf, (short)0, acc2[t2], false, false);
    }
  }

  // ---------------- order-1 GEMM: K = 16 -----------------------------------
  v8f acc1 = {};
  #pragma unroll
  for (int kc = 0; kc < 4; ++kc) {
    v2f bf;
    bf.x = lds_x[xbase + kc * 4 + h];
    bf.y = lds_x[xbase + kc * 4 + 2 + h];
    v2f af = *(const v2f*)&B1p[(kc * 2 + h) * (2 * R1) + col * 2];
    acc1 = __builtin_amdgcn_wmma_f32_16x16x4_f32(
        false, af, false, bf, (short)0, acc1, false, false);
  }

  // ---------------- weight contraction + reduction -------------------------
  // D-tile layout: VGPR v holds row (tile*16 + v + 8*h), column = lane%16.
  const int t = input_type[n];
  const int c = wv * 16 + col;
  float outv[9] = {};

  #pragma unroll
  for (int r = 0; r < 119; ++r) {
    const int rt = r >> 4, rm = r & 15, rv_ = rm & 7, hr = rm >> 3;
    const float* wp; int mul_, k_, gi;
    if (r < 8)       { wp = w30; mul_ = 8;  k_ = r;                gi = 0; }
    else if (r < 44) { int q = r - 8;  wp = w31; mul_ = 12; k_ = q / 3; gi = 1 + q % 3; }
    else             { int q = r - 44; wp = w32; mul_ = 15; k_ = q / 5; gi = 4 + q % 5; }
    const float wval = wp[((size_t)t * mul_ + k_) * 128 + c];
    outv[gi] += (h == hr) ? acc3[rt][rv_] * wval : 0.0f;
  }
  #pragma unroll
  for (int r = 0; r < 32; ++r) {
    const int rt = r >> 4, rm = r & 15, rv_ = rm & 7, hr = rm >> 3;
    const float* wp; int mul_, k_, gi;
    if (r < 3)       { wp = w20; mul_ = 3; k_ = r;                gi = 0; }
    else if (r < 12) { int q = r - 3;  wp = w21; mul_ = 3; k_ = q / 3; gi = 1 + q % 3; }
    else             { int q = r - 12; wp = w22; mul_ = 4; k_ = q / 5; gi = 4 + q % 5; }
    const float wval = wp[((size_t)t * mul_ + k_) * 128 + c];
    outv[gi] += (h == hr) ? acc2[rt][rv_] * wval : 0.0f;
  }
  #pragma unroll
  for (int r = 0; r < 9; ++r) {
    const int rv_ = r & 7, hr = r >> 3;
    const float* wp = (r < 1) ? w10 : (r < 4) ? w11 : w12;   // mul = 1
    const float wval = wp[(size_t)t * 128 + c];
    outv[r] += (h == hr) ? acc1[rv_] * wval : 0.0f;
  }

  #pragma unroll
  for (int i = 0; i < 9; ++i) outv[i] += __shfl_xor(outv[i], 16, 32);

  if (h == 0) {
    float* po = out + ((size_t)n * 128 + c) * 9;
    #pragma unroll
    for (int i = 0; i < 9; ++i) po[i] = outv[i];
  }
}

extern "C" void kernel_launch(void* const* d_in, const int* in_sizes, int n_in,
                              void* d_out, int out_size, void* d_ws, size_t ws_size,
                              hipStream_t stream) {
  (void)in_sizes; (void)n_in; (void)out_size; (void)ws_size;
  const float* x    = (const float*)d_in[0];
  const int*   ityp = (const int*)  d_in[1];
  const float* w30  = (const float*)d_in[2];
  const float* w31  = (const float*)d_in[3];
  const float* w32  = (const float*)d_in[4];
  const float* w20  = (const float*)d_in[5];
  const float* w21  = (const float*)d_in[6];
  const float* w22  = (const float*)d_in[7];
  const float* w10  = (const float*)d_in[8];
  const float* w11  = (const float*)d_in[9];
  const float* w12  = (const float*)d_in[10];
  const float* b30  = (const float*)d_in[11];
  const float* b31  = (const float*)d_in[12];
  const float* b32  = (const float*)d_in[13];
  const float* b20  = (const float*)d_in[14];
  const float* b21  = (const float*)d_in[15];
  const float* b22  = (const float*)d_in[16];
  const float* b10  = (const float*)d_in[17];
  const float* b11  = (const float*)d_in[18];
  const float* b12  = (const float*)d_in[19];
  float* ws  = (float*)d_ws;
  float* out = (float*)d_out;

  const int packTotal = D3K * R3 + D2K * R2 + D1K * R1;  // 532736 elems (~2.08 MB)
  pack_bases_kernel<<<(packTotal + 255) / 256, 256, 0, stream>>>(
      b30, b31, b32, b20, b21, b22, b10, b11, b12, ws);

  mace_symcon_kernel<<<1024, 256, 0, stream>>>(
      x, ityp, w30, w31, w32, w20, w21, w22, w10, w11, w12, ws, out);
}